// CustomAttention_90228672954384
// MI455X (gfx1250) — compile-verified
//
#include <hip/hip_runtime.h>
#include <math.h>

typedef float v2f __attribute__((ext_vector_type(2)));
typedef float v8f __attribute__((ext_vector_type(8)));

#define EMBED            2048
#define BATCH            32
#define BLOCK_THREADS    256
#define WAVES_PER_BLOCK  8          // 256 / wave32
#define ROWS_PER_WAVE    16
#define ROWS_PER_BLOCK   128        // 8 waves * 16 rows
#define BLOCKS_PER_BATCH 16         // 2048 / 128

__global__ __launch_bounds__(BLOCK_THREADS)
void attn_outer_softmax_wmma(const float* __restrict__ q,
                             const float* __restrict__ k,
                             const float* __restrict__ v,
                             float* __restrict__ out)
{
    __shared__ float sK[EMBED];                     // 8 KB
    __shared__ float sV[EMBED];                     // 8 KB
    __shared__ float sRed[2 * BLOCK_THREADS];       // 2 KB (kmax/kmin reduce)
    __shared__ float sOut[WAVES_PER_BLOCK][32];     // per-wave num/den scratch

    const int b    = blockIdx.y;
    const int tid  = threadIdx.x;
    const int wave = tid >> 5;
    const int lane = tid & 31;

    const float* kb = k + b * EMBED;
    const float* vb = v + b * EMBED;

    // ---- Stage K and V for this batch into LDS (reused 2048x) ----
    for (int i = tid; i < EMBED; i += BLOCK_THREADS) {
        sK[i] = kb[i];
        sV[i] = vb[i];
    }
    __syncthreads();

    // ---- Block-wide kmax / kmin (exact softmax max-subtraction) ----
    float kmx = -INFINITY, kmn = INFINITY;
    for (int i = tid; i < EMBED; i += BLOCK_THREADS) {
        float x = sK[i];
        kmx = fmaxf(kmx, x);
        kmn = fminf(kmn, x);
    }
    sRed[tid]                 = kmx;
    sRed[BLOCK_THREADS + tid] = kmn;
    __syncthreads();
    for (int s = BLOCK_THREADS / 2; s > 0; s >>= 1) {
        if (tid < s) {
            sRed[tid]                 = fmaxf(sRed[tid], sRed[tid + s]);
            sRed[BLOCK_THREADS + tid] = fminf(sRed[BLOCK_THREADS + tid],
                                              sRed[BLOCK_THREADS + tid + s]);
        }
        __syncthreads();
    }
    const float kmax = sRed[0];
    const float kmin = sRed[BLOCK_THREADS];

    // ---- Per-wave 16-row tile ----
    const int rowBase = blockIdx.x * ROWS_PER_BLOCK + wave * ROWS_PER_WAVE;
    const int n       = lane & 15;   // A: row index; B/C/D: column index
    const int h       = lane >> 4;   // K-half selector (K 0..1 vs 2..3)

    // Fold 1/sqrt(2048) AND log2(e) into the per-row scalar once:
    //   exp(t*k - m) == exp2(tl*k - m2),  tl = t*log2e, m2 = m*log2e
    const float tlScale = 0.02209708691207961f * 1.4426950408889634f;
    const float tl = q[b * EMBED + rowBase + n] * tlScale;
    // max_j (tl * k_j) = max(tl*kmax, tl*kmin)   (log2e > 0 preserves it)
    const float m2 = fmaxf(tl * kmax, tl * kmin);

    const bool isV   = (n == 0);   // B column 0 carries v_j  -> numerator
    const bool isOne = (n == 1);   // B column 1 carries 1.0  -> denominator

    v8f acc = {};
    #pragma unroll 8
    for (int j0 = 0; j0 < EMBED; j0 += 4) {
        // Lane's K-pair: lanes 0-15 take K={0,1}, lanes 16-31 take K={2,3}
        const float2 kk = *(const float2*)(&sK[j0 + 2 * h]);
        const float2 vv = *(const float2*)(&sV[j0 + 2 * h]);

        v2f a, bb;
        a.x = __builtin_amdgcn_exp2f(fmaf(tl, kk.x, -m2));
        a.y = __builtin_amdgcn_exp2f(fmaf(tl, kk.y, -m2));
        bb.x = isV ? vv.x : (isOne ? 1.0f : 0.0f);
        bb.y = isV ? vv.y : (isOne ? 1.0f : 0.0f);

        // D(16x16,f32) += A(16x4,f32) * B(4x16,f32)
        // col0 of D = running numerator, col1 = running denominator
        acc = __builtin_amdgcn_wmma_f32_16x16x4_f32(
            /*neg_a=*/false, a, /*neg_b=*/false, bb,
            /*c_mod=*/(short)0, acc, /*reuse_a=*/false, /*reuse_b=*/false);
    }

    // ---- Extract columns 0 and 1 of D via per-wave LDS scratch ----
    // D layout: VGPR r -> lanes 0-15 hold (M=r, N=lane), lanes 16-31 hold (M=8+r).
    if (n < 2) {
        #pragma unroll
        for (int r = 0; r < 8; ++r)
            sOut[wave][n * 16 + h * 8 + r] = acc[r];
    }
    __syncthreads();

    if (lane < 16) {
        const float num = sOut[wave][lane];
        const float den = sOut[wave][16 + lane];
        out[b * EMBED + rowBase + lane] = num / den;
    }
}

extern "C" void kernel_launch(void* const* d_in, const int* in_sizes, int n_in,
                              void* d_out, int out_size, void* d_ws, size_t ws_size,
                              hipStream_t stream) {
    const float* q = (const float*)d_in[0];
    const float* k = (const float*)d_in[1];
    const float* v = (const float*)d_in[2];
    float* out = (float*)d_out;

    dim3 grid(BLOCKS_PER_BATCH, BATCH);
    dim3 block(BLOCK_THREADS);
    attn_outer_softmax_wmma<<<grid, block, 0, stream>>>(q, k, v, out);
}